// CBPA_120259084711
// MI455X (gfx1250) — compile-verified
//
#include <hip/hip_runtime.h>
#include <hip/hip_bf16.h>

typedef __attribute__((ext_vector_type(2))) float v2f;
typedef __attribute__((ext_vector_type(8))) float v8f;

#define TPB 256

// ---------------- elementwise / graph kernels ----------------

__global__ void k_zero(float* __restrict__ p, int n) {
  int i = blockIdx.x * blockDim.x + threadIdx.x;
  if (i < n) p[i] = 0.0f;
}

__global__ void k_degree(const int* __restrict__ e, int E, int Nu,
                         float* __restrict__ deg) {
  int i = blockIdx.x * blockDim.x + threadIdx.x;
  if (i >= E) return;
  int u = e[2 * i + 0];
  int v = Nu + e[2 * i + 1];
  atomicAdd(&deg[u], 1.0f);
  atomicAdd(&deg[v], 1.0f);
}

__global__ void k_dinv(const float* __restrict__ deg, float* __restrict__ dinv, int n) {
  int i = blockIdx.x * blockDim.x + threadIdx.x;
  if (i >= n) return;
  float d = deg[i];
  dinv[i] = (d > 0.0f) ? rsqrtf(fmaxf(d, 1.0f)) : 0.0f;
}

// Symmetric LightGCN scatter: 8 threads per undirected edge, 4 dims each,
// both directions. Embedding table (19.2MB) is L2-resident on MI455X, so the
// float atomics hit the 192MB L2 atomic units.
__global__ void k_scatter(const int* __restrict__ e, int E, int Nu,
                          const float* __restrict__ dinv,
                          const float* __restrict__ cur,
                          float* __restrict__ nxt) {
  int t = blockIdx.x * blockDim.x + threadIdx.x;
  int total = E * 8;
  if (t >= total) return;
  int ei = t >> 3;
  int d  = (t & 7) << 2;
  int u = e[2 * ei + 0];
  int v = Nu + e[2 * ei + 1];
  float w = dinv[u] * dinv[v];
  const float4 cu = *(const float4*)(cur + (size_t)u * 32 + d);
  const float4 cv = *(const float4*)(cur + (size_t)v * 32 + d);
  float* nu = nxt + (size_t)u * 32 + d;
  float* nv = nxt + (size_t)v * 32 + d;
  atomicAdd(nv + 0, w * cu.x); atomicAdd(nv + 1, w * cu.y);
  atomicAdd(nv + 2, w * cu.z); atomicAdd(nv + 3, w * cu.w);
  atomicAdd(nu + 0, w * cv.x); atomicAdd(nu + 1, w * cv.y);
  atomicAdd(nu + 2, w * cv.z); atomicAdd(nu + 3, w * cv.w);
}

__global__ void k_axpy(float* __restrict__ acc, const float* __restrict__ x, int n) {
  int i = blockIdx.x * blockDim.x + threadIdx.x;
  if (i < n) acc[i] += x[i];
}

__global__ void k_scale(float* __restrict__ p, float s, int n) {
  int i = blockIdx.x * blockDim.x + threadIdx.x;
  if (i < n) p[i] *= s;
}

// ---------------- WMMA batched-dot kernel ----------------
// Each wave32 computes 16 dot products of 32-dim rows via 8x
// V_WMMA_F32_16X16X4_F32 accumulations; dots are the diagonal of the
// 16x16 f32 accumulator tile, extracted through LDS.
// A (16x4 f32): lanes 0-15 hold M=lane {K0,K1}; lanes 16-31 hold M=lane-16 {K2,K3}.
// B (4x16 f32): lanes 0-15 hold N=lane {K0,K1}; lanes 16-31 hold N=lane-16 {K2,K3}.
__global__ void k_dot_wmma(const float* __restrict__ emb, const int* __restrict__ bd,
                           int u_sub, int i_sub, int item_base, int do_relu,
                           float* __restrict__ out) {
  __shared__ float sm[4][16 * 17];
  int wid  = threadIdx.x >> 5;
  int lane = threadIdx.x & 31;
  int half = lane >> 4;
  int l16  = lane & 15;
  int tile0 = (blockIdx.x * (blockDim.x >> 5) + wid) << 4;
  int row = tile0 + l16;

  int uidx = bd[row * 9 + u_sub];
  int iidx = item_base + bd[row * 9 + i_sub];
  const float* ar = emb + (size_t)uidx * 32;
  const float* br = emb + (size_t)iidx * 32;

  v8f c = {};
#pragma unroll
  for (int k0 = 0; k0 < 32; k0 += 4) {
    v2f a, b;
    a.x = ar[k0 + half * 2 + 0];
    a.y = ar[k0 + half * 2 + 1];
    b.x = br[k0 + half * 2 + 0];
    b.y = br[k0 + half * 2 + 1];
    c = __builtin_amdgcn_wmma_f32_16x16x4_f32(false, a, false, b, (short)0, c,
                                              false, false);
  }

  // C/D layout: lanes 0-15: VGPR r -> (M=r,   N=lane)
  //             lanes16-31: VGPR r -> (M=8+r, N=lane-16)
  int mbase = half * 8;
#pragma unroll
  for (int r = 0; r < 8; ++r)
    sm[wid][(mbase + r) * 17 + l16] = c[r];
  __syncthreads();

  if (lane < 16) {
    float dv = sm[wid][l16 * 17 + l16];
    if (do_relu) dv = fmaxf(dv, 0.0f);
    out[row] = dv;
  }
}

// ---------------- reductions ----------------

__global__ void k_sumsq(const float* __restrict__ x, int n, float* __restrict__ part) {
  __shared__ float sm[TPB];
  float s = 0.0f;
  for (int i = blockIdx.x * blockDim.x + threadIdx.x; i < n;
       i += gridDim.x * blockDim.x) {
    float v = x[i];
    s += v * v;
  }
  sm[threadIdx.x] = s;
  __syncthreads();
  for (int o = TPB / 2; o > 0; o >>= 1) {
    if (threadIdx.x < o) sm[threadIdx.x] += sm[threadIdx.x + o];
    __syncthreads();
  }
  if (threadIdx.x == 0) part[blockIdx.x] = sm[0];
}

__device__ __forceinline__ float log_sigmoid(float x) {
  return fminf(x, 0.0f) - log1pf(expf(-fabsf(x)));
}

// scores layout: sc[(idx*8 + slot)*B], slots:
// 0=cps 1=cns 2=aps 3=ans 4=dps 5=dns 6=aux_ps 7=aux_ns
__global__ void k_final(const float* __restrict__ sc, const float* __restrict__ pu,
                        const float* __restrict__ pi, int npart, int B,
                        float itemRows, float* __restrict__ out) {
  __shared__ float smr[TPB], sm0[TPB], sm1[TPB];
  const float* cps0 = sc + 0 * B;  const float* cns0 = sc + 1 * B;
  const float* aps0 = sc + 2 * B;  const float* ans0 = sc + 3 * B;
  const float* dps0 = sc + 4 * B;  const float* dns0 = sc + 5 * B;
  const float* xps0 = sc + 6 * B;  const float* xns0 = sc + 7 * B;
  const float* cps1 = sc + 8 * B;  const float* cns1 = sc + 9 * B;
  const float* aps1 = sc + 10 * B; const float* ans1 = sc + 11 * B;
  const float* dps1 = sc + 12 * B; const float* dns1 = sc + 13 * B;
  const float* xps1 = sc + 14 * B; const float* xns1 = sc + 15 * B;

  float sr = 0.0f, s0 = 0.0f, s1 = 0.0f;
  for (int b = threadIdx.x; b < B; b += blockDim.x) {
    // front_door: p = (dps0+dps1) * (cps0*aps0 + cps1*aps1)
    float sp = cps0[b] * aps0[b] + cps1[b] * aps1[b];
    float sn = cns0[b] * ans0[b] + cns1[b] * ans1[b];
    float p = (dps0[b] + dps1[b]) * sp;
    float n = (dns0[b] + dns1[b]) * sn;
    sr += log_sigmoid(p - n);
    s0 += log_sigmoid(xps0[b] - xns0[b]);
    s1 += log_sigmoid(xps1[b] - xns1[b]);
  }
  smr[threadIdx.x] = sr; sm0[threadIdx.x] = s0; sm1[threadIdx.x] = s1;
  __syncthreads();
  for (int o = TPB / 2; o > 0; o >>= 1) {
    if (threadIdx.x < o) {
      smr[threadIdx.x] += smr[threadIdx.x + o];
      sm0[threadIdx.x] += sm0[threadIdx.x + o];
      sm1[threadIdx.x] += sm1[threadIdx.x + o];
    }
    __syncthreads();
  }
  if (threadIdx.x == 0) {
    float su = 0.0f, si = 0.0f;
    for (int i = 0; i < npart; ++i) { su += pu[i]; si += pi[i]; }
    float invB = 1.0f / (float)B;
    float rec_loss = -smr[0] * invB;
    float aux_loss = 0.5f * ((-sm0[0] * invB) + (-sm1[0] * invB)); // mean over 2
    float emb_loss = (sqrtf(su) + sqrtf(si)) / itemRows;
    out[0] = rec_loss + 0.5f * aux_loss + 1e-4f * emb_loss;
  }
}

// ---------------- host orchestration ----------------

static inline dim3 g1(int n) { return dim3((n + TPB - 1) / TPB); }

extern "C" void kernel_launch(void* const* d_in, const int* in_sizes, int n_in,
                              void* d_out, int out_size, void* d_ws, size_t ws_size,
                              hipStream_t stream) {
  const float* user_emb = (const float*)d_in[0];
  const float* item_emb = (const float*)d_in[1];
  const int*   bd       = (const int*)d_in[2];
  const int*   auxAll   = (const int*)d_in[3];
  const int*   tgtE     = (const int*)d_in[4];

  const int Nu = in_sizes[0] / 32;      // 100001
  const int Ni = in_sizes[1] / 32;      // 50001
  const int N  = Nu + Ni;               // 150002
  const int B  = in_sizes[2] / 9;       // 4096
  const int E  = in_sizes[4] / 2;       // 1000000
  const int ND = N * 32;

  float* ws   = (float*)d_ws;
  float* allE = ws;                 // N*32  : concatenated embeddings (read-only)
  float* bufA = allE + ND;          // N*32  : layer buffer
  float* bufB = bufA + ND;          // N*32  : layer buffer
  float* acc  = bufB + ND;          // N*32  : acc for comb / de passes
  float* AE   = acc + ND;           // N*32  : aux-propagated embeddings (kept)
  float* deg  = AE + ND;            // N
  float* dnv  = deg + N;            // N
  float* tdnv = dnv + N;            // N     : target-graph dinv (reused)
  float* sc   = tdnv + N;           // 16*B  : score arrays
  float* pu   = sc + 16 * B;        // 256   : sumsq partials (user)
  float* pi   = pu + TPB;           // 256   : sumsq partials (item)

  const size_t NDb = (size_t)ND * sizeof(float);
  const int dotBlocks = B / 64;     // 16 pairs/wave * 4 waves/block

  // all_emb = concat(user_emb, item_emb)
  hipMemcpyAsync(allE, user_emb, (size_t)Nu * 32 * sizeof(float),
                 hipMemcpyDeviceToDevice, stream);
  hipMemcpyAsync(allE + (size_t)Nu * 32, item_emb, (size_t)Ni * 32 * sizeof(float),
                 hipMemcpyDeviceToDevice, stream);

  // target-graph dinv (shared across both aux indices)
  k_zero<<<g1(N), TPB, 0, stream>>>(deg, N);
  k_degree<<<g1(E), TPB, 0, stream>>>(tgtE, E, Nu, deg);
  k_dinv<<<g1(N), TPB, 0, stream>>>(deg, tdnv, N);

  for (int idx = 0; idx < 2; ++idx) {
    const int* aE = auxAll + (size_t)idx * E * 2;
    float* S = sc + (size_t)idx * 8 * B;
    int rs = (1 + idx) * 3;  // aux row offset inside batch_data[b]

    // ---------- combined graph: ce = lightgcn(all_emb, comb) ----------
    k_zero<<<g1(N), TPB, 0, stream>>>(deg, N);
    k_degree<<<g1(E), TPB, 0, stream>>>(aE, E, Nu, deg);
    k_degree<<<g1(E), TPB, 0, stream>>>(tgtE, E, Nu, deg);
    k_dinv<<<g1(N), TPB, 0, stream>>>(deg, dnv, N);

    hipMemcpyAsync(acc, allE, NDb, hipMemcpyDeviceToDevice, stream);
    {
      const float* cur = allE;
      for (int L = 0; L < 2; ++L) {
        float* nxt = (L == 0) ? bufA : bufB;
        k_zero<<<g1(ND), TPB, 0, stream>>>(nxt, ND);
        k_scatter<<<g1(E * 8), TPB, 0, stream>>>(aE, E, Nu, dnv, cur, nxt);
        k_scatter<<<g1(E * 8), TPB, 0, stream>>>(tgtE, E, Nu, dnv, cur, nxt);
        k_axpy<<<g1(ND), TPB, 0, stream>>>(acc, nxt, ND);
        cur = nxt;
      }
    }
    k_scale<<<g1(ND), TPB, 0, stream>>>(acc, 1.0f / 3.0f, ND);
    k_dot_wmma<<<dotBlocks, 128, 0, stream>>>(acc, bd, 0, 1, Nu, 1, S + 0 * B);
    k_dot_wmma<<<dotBlocks, 128, 0, stream>>>(acc, bd, 0, 2, Nu, 1, S + 1 * B);

    // ---------- aux graph: ae = lightgcn(all_emb, aux) ----------
    k_zero<<<g1(N), TPB, 0, stream>>>(deg, N);
    k_degree<<<g1(E), TPB, 0, stream>>>(aE, E, Nu, deg);
    k_dinv<<<g1(N), TPB, 0, stream>>>(deg, dnv, N);

    hipMemcpyAsync(AE, allE, NDb, hipMemcpyDeviceToDevice, stream);
    {
      const float* cur = allE;
      for (int L = 0; L < 2; ++L) {
        float* nxt = (L == 0) ? bufA : bufB;
        k_zero<<<g1(ND), TPB, 0, stream>>>(nxt, ND);
        k_scatter<<<g1(E * 8), TPB, 0, stream>>>(aE, E, Nu, dnv, cur, nxt);
        k_axpy<<<g1(ND), TPB, 0, stream>>>(AE, nxt, ND);
        cur = nxt;
      }
    }
    k_scale<<<g1(ND), TPB, 0, stream>>>(AE, 1.0f / 3.0f, ND);
    k_dot_wmma<<<dotBlocks, 128, 0, stream>>>(AE, bd, 0, 1, Nu, 1, S + 2 * B);
    k_dot_wmma<<<dotBlocks, 128, 0, stream>>>(AE, bd, 0, 2, Nu, 1, S + 3 * B);
    k_dot_wmma<<<dotBlocks, 128, 0, stream>>>(AE, bd, rs + 0, rs + 1, Nu, 0, S + 6 * B);
    k_dot_wmma<<<dotBlocks, 128, 0, stream>>>(AE, bd, rs + 0, rs + 2, Nu, 0, S + 7 * B);

    // ---------- de = lightgcn(ae, target) ----------
    hipMemcpyAsync(acc, AE, NDb, hipMemcpyDeviceToDevice, stream);
    {
      const float* cur = AE;
      for (int L = 0; L < 2; ++L) {
        float* nxt = (L == 0) ? bufA : bufB;
        k_zero<<<g1(ND), TPB, 0, stream>>>(nxt, ND);
        k_scatter<<<g1(E * 8), TPB, 0, stream>>>(tgtE, E, Nu, tdnv, cur, nxt);
        k_axpy<<<g1(ND), TPB, 0, stream>>>(acc, nxt, ND);
        cur = nxt;
      }
    }
    k_scale<<<g1(ND), TPB, 0, stream>>>(acc, 1.0f / 3.0f, ND);
    k_dot_wmma<<<dotBlocks, 128, 0, stream>>>(acc, bd, 0, 1, Nu, 1, S + 4 * B);
    k_dot_wmma<<<dotBlocks, 128, 0, stream>>>(acc, bd, 0, 2, Nu, 1, S + 5 * B);
  }

  // embedding Frobenius norms
  k_sumsq<<<TPB, TPB, 0, stream>>>(user_emb, Nu * 32, pu);
  k_sumsq<<<TPB, TPB, 0, stream>>>(item_emb, Ni * 32, pi);

  // final scalar loss
  k_final<<<1, TPB, 0, stream>>>(sc, pu, pi, TPB, B, (float)Ni, (float*)d_out);
}